// RobustRnn_10608569221425
// MI455X (gfx1250) — compile-verified
//
#include <hip/hip_runtime.h>
#include <hip/hip_bf16.h>

typedef __attribute__((ext_vector_type(2))) float v2f;
typedef __attribute__((ext_vector_type(8))) float v8f;

namespace {
constexpr int kB    = 32;
constexpr int kNU   = 64;
constexpr int kNY   = 64;
constexpr int kNX   = 512;
constexpr int kT    = 1024;
constexpr int kLDW1 = 576;          // packed [Cv|Dv ; Fh|Bh2] row length
constexpr int kNWG  = 16;           // persistent workgroups
constexpr int kWaves = kNWG * 8;    // 128 wave32 waves

// workspace layout (float offsets)
constexpr size_t O_X      = 256;                         // Newton buf A (512x512)
constexpr size_t O_X2     = O_X  + 512 * 512;            // Newton buf B
constexpr size_t O_T1     = O_X2 + 512 * 512;            // Newton temp
constexpr size_t O_W1     = O_T1 + 512 * 512;            // 1024 x 576 fused weights
constexpr size_t O_BH1    = O_W1 + (size_t)1024 * kLDW1; // Einv@B1_w (512x512)
constexpr size_t O_BH     = O_BH1 + 512 * 512;           // Einv@F_b (512)
constexpr size_t O_P      = O_BH + 512;                  // per-step 32x1024 staging
constexpr size_t O_STATES = O_P + (size_t)32 * 1024;     // states (B,T,NX) 64MB
} // namespace

__device__ __forceinline__ v8f vzero() {
  v8f c;
#pragma unroll
  for (int i = 0; i < 8; ++i) c[i] = 0.0f;
  return c;
}

__device__ __forceinline__ v8f wmma4(v2f a, v2f b, v8f c) {
  // D = A(16x4) * B(4x16) + C, fp32 everywhere
  return __builtin_amdgcn_wmma_f32_16x16x4_f32(false, a, false, b, (short)0, c,
                                               false, false);
}

// C/D tile layout: element (M = v + 8*(lane>>4), N = lane&15)
__device__ __forceinline__ void store_tile(float* dst, int ldd, const v8f& c,
                                           int lane) {
  const int N  = lane & 15;
  const int Mb = (lane >> 4) * 8;
#pragma unroll
  for (int v = 0; v < 8; ++v) dst[(size_t)(Mb + v) * ldd + N] = c[v];
}

__device__ __forceinline__ void grid_barrier(unsigned* cnt, unsigned& phase) {
  __syncthreads();
  ++phase;
  if (threadIdx.x == 0) {
    __threadfence();
    atomicAdd(cnt, 1u);
    const unsigned target = phase * (unsigned)kNWG;
    while (atomicAdd(cnt, 0u) < target) __builtin_amdgcn_s_sleep(2);
  }
  __syncthreads();
  __threadfence();
}

__global__ void rnn_init_kernel(unsigned* cnt) {
  if (threadIdx.x == 0) atomicExch(cnt, 0u);
}

// One persistent kernel: Newton inverse of E, fused-weight prep, then the
// 1023-step sequential scan with two WMMA GEMM phases per step.
__global__ __launch_bounds__(256) void rnn_scan_kernel(
    const float* __restrict__ u, const float* __restrict__ E,
    const float* __restrict__ Fw, const float* __restrict__ Fb,
    const float* __restrict__ B1w, const float* __restrict__ B2w,
    const float* __restrict__ C2t, const float* __restrict__ bv,
    const float* __restrict__ Dt, const float* __restrict__ multis,
    float* __restrict__ ws) {
  constexpr int LDS_A = 516;  // 516 % 64 = 4 -> conflict-free across 16 rows
  constexpr int LDS_U = 68;
  __shared__ float Alds[16 * LDS_A];   // staged 16x512 A-block (h_t / relu(V))
  __shared__ float Uldss[16 * LDS_U];  // staged 16x64 u_t block
  __shared__ float Pred[4 * 256];      // phase-2 split-K partial sums

  unsigned* cnt = (unsigned*)ws;
  float* Xa  = ws + O_X;
  float* Xb  = ws + O_X2;
  float* T1  = ws + O_T1;
  float* W1  = ws + O_W1;
  float* Bh1 = ws + O_BH1;
  float* bh  = ws + O_BH;
  float* P   = ws + O_P;
  float* st  = ws + O_STATES;

  const int tid  = blockIdx.x * 256 + threadIdx.x;  // 0..4095
  const int wv   = tid >> 5;                        // 0..127
  const int lw   = threadIdx.x >> 5;                // local wave 0..7
  const int lane = threadIdx.x & 31;
  const int r    = lane & 15;
  const int kh   = (lane >> 4) * 2;
  const int N    = lane & 15;
  const int Mb   = (lane >> 4) * 8;
  const int m0blk = (blockIdx.x >= 8) ? 16 : 0;     // == (wv>>6)*16, block-uniform
  unsigned phase = 0;

  // ---------- init: X0 = I, W1 top half = [Cv|Dv], h0 = 0 ----------
  for (int i = tid; i < 512 * 512; i += kNWG * 256)
    Xa[i] = ((i >> 9) == (i & 511)) ? 1.0f : 0.0f;
  for (int i = tid; i < 512 * kLDW1; i += kNWG * 256) {
    const int n = i / kLDW1, k = i - n * kLDW1;
    const float inv_m = 1.0f / multis[n];
    W1[i] = (k < 512 ? C2t[(size_t)n * 512 + k] : Dt[(size_t)n * 64 + (k - 512)]) * inv_m;
  }
  for (int i = tid; i < kB * kNX; i += kNWG * 256) {
    const int b = i >> 9, x = i & 511;
    st[(size_t)b * kT * kNX + x] = 0.0f;  // states[:,0,:] = 0
  }
  grid_barrier(cnt, phase);

  // ---------- Newton iteration: X <- X(2I - E X), 7 iters ----------
  float* cur = Xa;
  float* nxt = Xb;
  for (int it = 0; it < 7; ++it) {
    // T1 = E @ cur
    for (int tile = wv; tile < 1024; tile += kWaves) {
      const int m0 = (tile >> 5) * 16, n0 = (tile & 31) * 16;
      v8f c = vzero();
      const float* ap = E + (size_t)(m0 + r) * 512 + kh;
      const float* bp = cur + (size_t)kh * 512 + n0 + r;
#pragma unroll 4
      for (int k = 0; k < 512; k += 4) {
        v2f a = *(const v2f*)(ap + k);
        v2f b;
        b.x = bp[(size_t)k * 512];
        b.y = bp[(size_t)k * 512 + 512];
        c = wmma4(a, b, c);
      }
      store_tile(T1 + (size_t)m0 * 512 + n0, 512, c, lane);
    }
    grid_barrier(cnt, phase);
    // nxt = 2*cur - cur @ T1
    for (int tile = wv; tile < 1024; tile += kWaves) {
      const int m0 = (tile >> 5) * 16, n0 = (tile & 31) * 16;
      v8f c = vzero();
      const float* ap = cur + (size_t)(m0 + r) * 512 + kh;
      const float* bp = T1 + (size_t)kh * 512 + n0 + r;
#pragma unroll 4
      for (int k = 0; k < 512; k += 4) {
        v2f a = *(const v2f*)(ap + k);
        v2f b;
        b.x = bp[(size_t)k * 512];
        b.y = bp[(size_t)k * 512 + 512];
        c = wmma4(a, b, c);
      }
#pragma unroll
      for (int v = 0; v < 8; ++v) {
        const size_t idx = (size_t)(m0 + Mb + v) * 512 + n0 + N;
        nxt[idx] = 2.0f * cur[idx] - c[v];
      }
    }
    grid_barrier(cnt, phase);
    float* t = cur; cur = nxt; nxt = t;
  }
  // cur == Einv now

  // ---------- fused weights: Fh, Bh1, Bh2 = Einv @ {F_w, B1_w, B2_w} ----------
  for (int tile = wv; tile < 2176; tile += kWaves) {
    int m0, n0, ldb, ldd;
    const float* Bsrc;
    float* dst;
    if (tile < 1024) {
      m0 = (tile >> 5) * 16; n0 = (tile & 31) * 16;
      Bsrc = Fw; ldb = 512; dst = W1 + (size_t)(512 + m0) * kLDW1 + n0; ldd = kLDW1;
    } else if (tile < 2048) {
      const int q = tile - 1024;
      m0 = (q >> 5) * 16; n0 = (q & 31) * 16;
      Bsrc = B1w; ldb = 512; dst = Bh1 + (size_t)m0 * 512 + n0; ldd = 512;
    } else {
      const int q = tile - 2048;
      m0 = (q >> 2) * 16; n0 = (q & 3) * 16;
      Bsrc = B2w; ldb = 64; dst = W1 + (size_t)(512 + m0) * kLDW1 + 512 + n0; ldd = kLDW1;
    }
    v8f c = vzero();
    const float* ap = cur + (size_t)(m0 + r) * 512 + kh;
    const float* bp = Bsrc + (size_t)kh * ldb + n0 + r;
#pragma unroll 4
    for (int k = 0; k < 512; k += 4) {
      v2f a = *(const v2f*)(ap + k);
      v2f b;
      b.x = bp[(size_t)k * ldb];
      b.y = bp[(size_t)k * ldb + ldb];
      c = wmma4(a, b, c);
    }
    store_tile(dst, ldd, c, lane);
  }
  // bh = Einv @ F_b
  for (int n = tid; n < 512; n += kNWG * 256) {
    float s = 0.0f;
    for (int j = 0; j < 512; ++j) s += cur[(size_t)n * 512 + j] * Fb[j];
    bh[n] = s;
  }
  grid_barrier(cnt, phase);

  // ---------- sequential scan: 1023 steps ----------
  for (int t = 0; t < kT - 1; ++t) {
    // stage A-block (h_t rows m0blk..m0blk+15) and u_t block into LDS
    for (int i = threadIdx.x; i < 16 * 128; i += 256) {
      const int row = i >> 7, x4 = (i & 127) * 4;
      *(float4*)(Alds + row * LDS_A + x4) =
          *(const float4*)(st + (size_t)(m0blk + row) * kT * kNX +
                           (size_t)t * kNX + x4);
    }
    for (int i = threadIdx.x; i < 16 * 64; i += 256) {
      const int row = i >> 6, x = i & 63;
      Uldss[row * LDS_U + x] =
          u[(size_t)(m0blk + row) * kNU * kT + (size_t)x * kT + t];
    }
    __syncthreads();

    // phase 1: P(32x1024) = [h_t | u_t] @ W1^T ; relu+bv left half, +bh right
    {
      const int n0 = (wv & 63) * 16;       // 64 col tiles, m-tile is block-uniform
      v8f c = vzero();
      const float* sp = Alds + r * LDS_A + kh;
      const float* wp = W1 + (size_t)(n0 + r) * kLDW1 + kh;
#pragma unroll 8
      for (int k = 0; k < 512; k += 4) {
        v2f a = *(const v2f*)(sp + k);
        v2f b = *(const v2f*)(wp + k);
        c = wmma4(a, b, c);
      }
      const float* upd = Uldss + r * LDS_U + kh;
#pragma unroll 4
      for (int k = 0; k < 64; k += 4) {
        v2f a = *(const v2f*)(upd + k);
        v2f b = *(const v2f*)(wp + 512 + k);
        c = wmma4(a, b, c);
      }
      if (n0 < 512) {
        const float sb = bv[n0 + N];
#pragma unroll
        for (int v = 0; v < 8; ++v)
          P[(size_t)(m0blk + Mb + v) * 1024 + n0 + N] = fmaxf(c[v] + sb, 0.0f);
      } else {
        const float sb = bh[n0 - 512 + N];
#pragma unroll
        for (int v = 0; v < 8; ++v)
          P[(size_t)(m0blk + Mb + v) * 1024 + n0 + N] = c[v] + sb;
      }
    }
    grid_barrier(cnt, phase);

    // stage relu(V) rows m0blk..m0blk+15 (P cols 0..511) into LDS
    for (int i = threadIdx.x; i < 16 * 128; i += 256) {
      const int row = i >> 7, x4 = (i & 127) * 4;
      *(float4*)(Alds + row * LDS_A + x4) =
          *(const float4*)(P + (size_t)(m0blk + row) * 1024 + x4);
    }
    __syncthreads();

    // phase 2: h_{t+1} = relu(V) @ Bh1^T + Hpart, K split across wave pairs
    {
      const int tt   = wv >> 1;            // tile 0..63 (m-tile == m0blk)
      const int n0   = (tt & 31) * 16;
      const int odd  = lw & 1;
      const int koff = odd * 256;
      const int pair = lw >> 1;            // 0..3
      v8f c = vzero();
      const float* ap = Alds + r * LDS_A + kh;
      const float* bp = Bh1 + (size_t)(n0 + r) * kNX + kh;
#pragma unroll 8
      for (int k = koff; k < koff + 256; k += 4) {
        v2f a = *(const v2f*)(ap + k);
        v2f b = *(const v2f*)(bp + k);
        c = wmma4(a, b, c);
      }
      if (odd) {
#pragma unroll
        for (int v = 0; v < 8; ++v) Pred[pair * 256 + v * 32 + lane] = c[v];
      }
      __syncthreads();
      if (!odd) {
#pragma unroll
        for (int v = 0; v < 8; ++v) {
          const int row = m0blk + Mb + v;
          const float hp = P[(size_t)row * 1024 + 512 + n0 + N];
          st[(size_t)row * kT * kNX + (size_t)(t + 1) * kNX + n0 + N] =
              c[v] + Pred[pair * 256 + v * 32 + lane] + hp;
        }
      }
    }
    grid_barrier(cnt, phase);
  }
}

// Parallel output stage: per 16-timestep block, stage S/U in LDS, compute
// W = relu(S@Cv^T + U@Dv^T + bv) into LDS, then Y = S@C1^T + W@D11^T + U@D12^T + by,
// stored transposed to (B, NY, T).
__global__ __launch_bounds__(256) void rnn_output_kernel(
    const float* __restrict__ u, const float* __restrict__ C1w,
    const float* __restrict__ D11w, const float* __restrict__ D12w,
    const float* __restrict__ by, const float* __restrict__ bv,
    const float* __restrict__ W1, const float* __restrict__ st,
    float* __restrict__ out) {
  constexpr int LDS_S = 516;  // pad to avoid bank conflicts (516 % 64 = 4)
  constexpr int LDS_U = 68;
  __shared__ float Slds[16 * LDS_S];
  __shared__ float Wlds[16 * LDS_S];
  __shared__ float Ulds[16 * LDS_U];

  const int b  = blockIdx.x >> 6;          // 64 blocks per batch row
  const int t0 = (blockIdx.x & 63) * 16;
  const int tidb = threadIdx.x;
  const int wv = tidb >> 5, lane = tidb & 31;
  const int r = lane & 15, kh = (lane >> 4) * 2;
  const int N = lane & 15, Mb = (lane >> 4) * 8;

  for (int i = tidb; i < 16 * 128; i += 256) {
    const int row = i >> 7, x4 = (i & 127) * 4;
    *(float4*)(Slds + row * LDS_S + x4) =
        *(const float4*)(st + (size_t)b * kT * kNX +
                         (size_t)(t0 + row) * kNX + x4);
  }
  for (int i = tidb; i < 16 * 64; i += 256) {
    const int row = i >> 6, x = i & 63;
    Ulds[row * LDS_U + x] = u[(size_t)b * kNU * kT + (size_t)x * kT + (t0 + row)];
  }
  __syncthreads();

  // V/W stage: 32 column tiles of 16, 4 per wave
  for (int j = 0; j < 4; ++j) {
    const int n0 = (wv * 4 + j) * 16;
    v8f c = vzero();
    const float* sp = Slds + r * LDS_S + kh;
    const float* wp = W1 + (size_t)(n0 + r) * kLDW1 + kh;
#pragma unroll 8
    for (int k = 0; k < 512; k += 4) {
      v2f a = *(const v2f*)(sp + k);
      v2f b = *(const v2f*)(wp + k);
      c = wmma4(a, b, c);
    }
    const float* upd = Ulds + r * LDS_U + kh;
#pragma unroll 4
    for (int k = 0; k < 64; k += 4) {
      v2f a = *(const v2f*)(upd + k);
      v2f b = *(const v2f*)(wp + 512 + k);
      c = wmma4(a, b, c);
    }
    const float sb = bv[n0 + N];
#pragma unroll
    for (int v = 0; v < 8; ++v)
      Wlds[(Mb + v) * LDS_S + n0 + N] = fmaxf(c[v] + sb, 0.0f);
  }
  __syncthreads();

  // Y stage: NY = 64 -> 4 column tiles on waves 0..3
  if (wv < 4) {
    const int n0 = wv * 16;
    v8f c = vzero();
    const float* sp  = Slds + r * LDS_S + kh;
    const float* wlp = Wlds + r * LDS_S + kh;
    const float* upd = Ulds + r * LDS_U + kh;
    const float* c1p  = C1w + (size_t)(n0 + r) * kNX + kh;
    const float* d11p = D11w + (size_t)(n0 + r) * 512 + kh;
    const float* d12p = D12w + (size_t)(n0 + r) * 64 + kh;
#pragma unroll 8
    for (int k = 0; k < 512; k += 4) {
      v2f a = *(const v2f*)(sp + k);
      v2f b = *(const v2f*)(c1p + k);
      c = wmma4(a, b, c);
    }
#pragma unroll 8
    for (int k = 0; k < 512; k += 4) {
      v2f a = *(const v2f*)(wlp + k);
      v2f b = *(const v2f*)(d11p + k);
      c = wmma4(a, b, c);
    }
#pragma unroll 4
    for (int k = 0; k < 64; k += 4) {
      v2f a = *(const v2f*)(upd + k);
      v2f b = *(const v2f*)(d12p + k);
      c = wmma4(a, b, c);
    }
    const float sb = by[n0 + N];
#pragma unroll
    for (int v = 0; v < 8; ++v)
      out[(size_t)b * kNY * kT + (size_t)(n0 + N) * kT + (t0 + Mb + v)] = c[v] + sb;
  }
}

extern "C" void kernel_launch(void* const* d_in, const int* in_sizes, int n_in,
                              void* d_out, int out_size, void* d_ws,
                              size_t ws_size, hipStream_t stream) {
  (void)in_sizes; (void)n_in; (void)out_size; (void)ws_size;
  const float* u      = (const float*)d_in[0];
  const float* E      = (const float*)d_in[1];
  const float* F_w    = (const float*)d_in[2];
  const float* F_b    = (const float*)d_in[3];
  const float* B1_w   = (const float*)d_in[4];
  const float* B2_w   = (const float*)d_in[5];
  const float* C2tild = (const float*)d_in[6];
  const float* bv     = (const float*)d_in[7];
  const float* Dtild  = (const float*)d_in[8];
  const float* C1_w   = (const float*)d_in[9];
  const float* D11_w  = (const float*)d_in[10];
  const float* D12_w  = (const float*)d_in[11];
  const float* by     = (const float*)d_in[12];
  const float* multis = (const float*)d_in[13];
  float* ws  = (float*)d_ws;
  float* out = (float*)d_out;

  rnn_init_kernel<<<1, 64, 0, stream>>>((unsigned*)d_ws);
  rnn_scan_kernel<<<kNWG, 256, 0, stream>>>(u, E, F_w, F_b, B1_w, B2_w, C2tild,
                                            bv, Dtild, multis, ws);
  rnn_output_kernel<<<kB * (kT / 16), 256, 0, stream>>>(
      u, C1_w, D11_w, D12_w, by, bv, ws + O_W1, ws + O_STATES, out);
}